// TensorProduct_66288525246799
// MI455X (gfx1250) — compile-verified
//
#include <hip/hip_runtime.h>
#include <hip/hip_bf16.h>

// e3nn-style channelwise tensor product for MI455X (gfx1250, wave32).
// Strategy: per-wave 16-row M-tiles, V_WMMA_F32_16X16X4_F32 for all five
// skinny GEMMs, x1 tile staged in LDS (padded stride 324 -> bank-conflict-free
// strided A reads), weights (152 KB total) read from global (L2/WGP$-resident).

typedef float v2f __attribute__((ext_vector_type(2)));
typedef float v8f __attribute__((ext_vector_type(8)));

#define MULS 128
#define MULV 64
#define DIM1 320          // 128 + 3*64
#define NOUT 416          // 128 + 192 + 96
#define ROWSTR 324        // padded LDS row stride (floats); 324 % 64 == 4
#define WAVES_PER_BLOCK 2
#define INV_SQRT3 0.57735026918962576f
#define INV_SQRT6 0.40824829046386302f

static __device__ __forceinline__ v8f wmma4(v2f a, v2f b, v8f c) {
  // D = A(16x4, f32) x B(4x16, f32) + C(16x16, f32)
  return __builtin_amdgcn_wmma_f32_16x16x4_f32(false, a, false, b, (short)0, c,
                                               false, false);
}

// B fragment: lane (hi half h, lane&15 = l) supplies B[k0+2h][n0+l] (VGPR0)
// and B[k0+2h+1][n0+l] (VGPR1). W is row-major KxN.
static __device__ __forceinline__ v2f ldb(const float* __restrict__ W, int N,
                                          int k0, int col, int hi2) {
  const float* p = W + (size_t)(k0 + hi2) * N + col;
  v2f b;
  b.x = p[0];
  b.y = p[N];
  return b;
}

__global__ __launch_bounds__(32 * WAVES_PER_BLOCK, 1)
void tp_wmma_kernel(const float* __restrict__ x1,
                    const float* __restrict__ x2,
                    const float* __restrict__ w000,   // 128x128
                    const float* __restrict__ w011,   // 128x64
                    const float* __restrict__ w101,   // 64x64
                    const float* __restrict__ w110,   // 64x128
                    const float* __restrict__ w112,   // 64x32
                    float* __restrict__ out,          // M x 416
                    int M, int ntiles) {
  __shared__ __align__(16) float lds[WAVES_PER_BLOCK * 16 * ROWSTR];

  const int lane = threadIdx.x & 31;
  const int wv   = threadIdx.x >> 5;
  const int tile = blockIdx.x * WAVES_PER_BLOCK + wv;
  if (tile >= ntiles) return;   // wave-uniform
  const int m0 = tile << 4;

  float* __restrict__ xt = lds + wv * (16 * ROWSTR);

  // ---- Stage x1[m0:m0+16, :] (16*320 contiguous floats) into LDS, padded rows.
  {
    const float4* __restrict__ src =
        reinterpret_cast<const float4*>(x1) + (size_t)m0 * (DIM1 / 4);
    if (m0 + 16 <= M) {
#pragma unroll
      for (int i = 0; i < 40; ++i) {
        const int idx = i * 32 + lane;
        const float4 v = src[idx];
        const int r = idx / 80, c4 = idx % 80;
        *reinterpret_cast<float4*>(xt + r * ROWSTR + c4 * 4) = v;
      }
    } else {  // tail tile: clamp reads, zero-fill
      const int rem = (M - m0) * (DIM1 / 4);
#pragma unroll
      for (int i = 0; i < 40; ++i) {
        const int idx = i * 32 + lane;
        float4 v = make_float4(0.f, 0.f, 0.f, 0.f);
        if (idx < rem) v = src[idx];
        const int r = idx / 80, c4 = idx % 80;
        *reinterpret_cast<float4*>(xt + r * ROWSTR + c4 * 4) = v;
      }
    }
  }

  // ---- Per-lane constants.
  const int rA  = lane & 15;   // A-matrix row owned by this lane
  const int hi  = lane >> 4;   // half: K offset 2*hi
  const int hi2 = hi * 2;
  const float* __restrict__ arow = xt + rA * ROWSTR;   // scalar features
  const float* __restrict__ vrow = arow + MULS;        // vector features (u,3)

  const float4* __restrict__ x2v4 = reinterpret_cast<const float4*>(x2);
  int mA = m0 + rA; if (mA > M - 1) mA = M - 1;
  const float4 xA = x2v4[mA];
  const float sA  = xA.x;                    // x2 scalar for this A row
  const float sA3 = xA.x * INV_SQRT3;
  const float y0 = xA.y, y1 = xA.z, y2 = xA.w;            // x2 vector
  const float y0s = y0 * INV_SQRT3, y1s = y1 * INV_SQRT3, y2s = y2 * INV_SQRT3;

  // x2 for the 8 C-rows this lane stores (row = v + 8*hi).
  float4 cr[8];
#pragma unroll
  for (int v = 0; v < 8; ++v) {
    int rc = m0 + v + hi * 8; if (rc > M - 1) rc = M - 1;
    cr[v] = x2v4[rc];
  }

  // ---- Cache scalar-feature A fragments (cols 0..127) in registers:
  // reused by 8 out0 N-tiles + 4 t011 N-tiles.
  float asc[64];
#pragma unroll
  for (int s = 0; s < 32; ++s) {
    asc[2 * s]     = arow[4 * s + hi2];
    asc[2 * s + 1] = arow[4 * s + hi2 + 1];
  }

  const v8f vzero = {0.f, 0.f, 0.f, 0.f, 0.f, 0.f, 0.f, 0.f};

  // ================= out0 (cols 0..127) =================
  // out0 = (x1s*x2s) @ W000 + (x1v . x2v/sqrt3) @ W110
#pragma unroll 1
  for (int nt = 0; nt < 8; ++nt) {
    const int col = nt * 16 + rA;
    v8f c = vzero;
#pragma unroll
    for (int s = 0; s < 32; ++s) {           // K = 128 over W000
      v2f a;
      a.x = asc[2 * s] * sA;
      a.y = asc[2 * s + 1] * sA;
      c = wmma4(a, ldb(w000, 128, 4 * s, col, hi2), c);
    }
#pragma unroll
    for (int s = 0; s < 16; ++s) {           // K = 64 over W110 (dvec on the fly)
      const float* p = vrow + 3 * (4 * s + hi2);
      v2f a;
      a.x = p[0] * y0s + p[1] * y1s + p[2] * y2s;
      a.y = p[3] * y0s + p[4] * y1s + p[5] * y2s;
      c = wmma4(a, ldb(w110, 128, 4 * s, col, hi2), c);
    }
#pragma unroll
    for (int v = 0; v < 8; ++v) {
      const int row = m0 + v + hi * 8;
      if (row < M) out[(size_t)row * NOUT + col] = c[v];
    }
  }

  // ================= out1 (cols 128..319) =================
  // out1[:,w,j] = (x1s @ W011)*x2v_j/sqrt3 + (x1v_j * x2s/sqrt3) @ W101
#pragma unroll 1
  for (int nt = 0; nt < 4; ++nt) {
    const int col = nt * 16 + rA;
    v8f t = vzero;
#pragma unroll
    for (int s = 0; s < 32; ++s) {           // t = x1s @ W011 (raw)
      v2f a;
      a.x = asc[2 * s];
      a.y = asc[2 * s + 1];
      t = wmma4(a, ldb(w011, 64, 4 * s, col, hi2), t);
    }
    v8f c0 = vzero, c1 = vzero, c2 = vzero;
#pragma unroll
    for (int s = 0; s < 16; ++s) {           // share B across j = 0,1,2
      const v2f b = ldb(w101, 64, 4 * s, col, hi2);
      const float* p = vrow + 3 * (4 * s + hi2);
      v2f a0, a1, a2;
      a0.x = p[0] * sA3;  a0.y = p[3] * sA3;
      a1.x = p[1] * sA3;  a1.y = p[4] * sA3;
      a2.x = p[2] * sA3;  a2.y = p[5] * sA3;
      c0 = wmma4(a0, b, c0);
      c1 = wmma4(a1, b, c1);
      c2 = wmma4(a2, b, c2);
    }
#pragma unroll
    for (int v = 0; v < 8; ++v) {
      const int row = m0 + v + hi * 8;
      if (row < M) {
        float* o = out + (size_t)row * NOUT + MULS + col * 3;
        o[0] = c0[v] + t[v] * (cr[v].y * INV_SQRT3);
        o[1] = c1[v] + t[v] * (cr[v].z * INV_SQRT3);
        o[2] = c2[v] + t[v] * (cr[v].w * INV_SQRT3);
      }
    }
  }

  // ================= out2 (cols 320..415) =================
  // out2[:,w,k] = ((x1v x x2v)_k @ W112) / sqrt6
#pragma unroll 1
  for (int nt = 0; nt < 2; ++nt) {
    const int col = nt * 16 + rA;
    v8f c0 = vzero, c1 = vzero, c2 = vzero;
#pragma unroll
    for (int s = 0; s < 16; ++s) {           // share B across k = 0,1,2
      const v2f b = ldb(w112, 32, 4 * s, col, hi2);
      const float* p = vrow + 3 * (4 * s + hi2);
      const float u0 = p[0], u1 = p[1], u2 = p[2];
      const float q0 = p[3], q1 = p[4], q2 = p[5];
      v2f a0, a1, a2;
      a0.x = u1 * y2 - u2 * y1;  a0.y = q1 * y2 - q2 * y1;
      a1.x = u2 * y0 - u0 * y2;  a1.y = q2 * y0 - q0 * y2;
      a2.x = u0 * y1 - u1 * y0;  a2.y = q0 * y1 - q1 * y0;
      c0 = wmma4(a0, b, c0);
      c1 = wmma4(a1, b, c1);
      c2 = wmma4(a2, b, c2);
    }
#pragma unroll
    for (int v = 0; v < 8; ++v) {
      const int row = m0 + v + hi * 8;
      if (row < M) {
        float* o = out + (size_t)row * NOUT + 320 + col * 3;
        o[0] = c0[v] * INV_SQRT6;
        o[1] = c1[v] * INV_SQRT6;
        o[2] = c2[v] * INV_SQRT6;
      }
    }
  }
}

extern "C" void kernel_launch(void* const* d_in, const int* in_sizes, int n_in,
                              void* d_out, int out_size, void* d_ws, size_t ws_size,
                              hipStream_t stream) {
  const float* x1   = (const float*)d_in[0];
  const float* x2   = (const float*)d_in[1];
  const float* w000 = (const float*)d_in[2];
  const float* w011 = (const float*)d_in[3];
  const float* w101 = (const float*)d_in[4];
  const float* w110 = (const float*)d_in[5];
  const float* w112 = (const float*)d_in[6];
  // d_in[7] = w3j111 (epsilon/sqrt6) is hardcoded as the cross product.
  float* out = (float*)d_out;

  const int M = in_sizes[0] / DIM1;
  const int ntiles = (M + 15) >> 4;
  const int blocks = (ntiles + WAVES_PER_BLOCK - 1) / WAVES_PER_BLOCK;

  tp_wmma_kernel<<<dim3(blocks), dim3(32 * WAVES_PER_BLOCK), 0, stream>>>(
      x1, x2, w000, w011, w101, w110, w112, out, M, ntiles);
}